// MinimalAttractorLM_22737556865290
// MI455X (gfx1250) — compile-verified
//
#include <hip/hip_runtime.h>

// ---------------------------------------------------------------------------
// MinimalAttractorLM for MI455X (gfx1250, wave32)
//
// Three phases:
//  1) precompute_kernel : xs[t] = embed[tok[t]],  cs[t] = W_b + xs[t] @ Wx_w.T
//  2) scan_kernel       : sequential 4096-step relaxation (1 wave, v_tanh_f32)
//  3) logits_kernel     : hs @ out_w.T + out_b via V_WMMA_F32_16X16X4_F32
//     store-bound: 823 MB fp32 out / 23.3 TB/s ~ 35 us floor. NT stores keep
//     out_w (3.2 MB) + hs (256 KB) L2-resident across the 16x N-tile reuse.
// ---------------------------------------------------------------------------

typedef __attribute__((ext_vector_type(2))) float v2f;
typedef __attribute__((ext_vector_type(8))) float v8f;

#define T_LEN  4096
#define D      16
#define VOCABN 50257
#define ALPHA  0.25f

// ---------------------------------------------------------------------------
// Kernel 1: per-token parallel precompute (trivial, 65536 threads)
// xs[t*16+i] = embed[tok[t]][i]
// cs[t*16+i] = W_b[i] + sum_j embed[tok[t]][j] * Wx_w[i][j]
// ---------------------------------------------------------------------------
__global__ void precompute_kernel(const int* __restrict__ tok,
                                  const float* __restrict__ embed,
                                  const float* __restrict__ Wx_w,
                                  const float* __restrict__ W_b,
                                  float* __restrict__ xs,
                                  float* __restrict__ cs) {
    int u = blockIdx.x * blockDim.x + threadIdx.x;
    if (u >= T_LEN * D) return;
    int t = u >> 4;
    int i = u & 15;
    int id = tok[t];
    const float* e = embed + (long long)id * D;
    xs[u] = e[i];
    float acc = W_b[i];
    const float* w = Wx_w + i * D;
#pragma unroll
    for (int j = 0; j < D; ++j) acc = fmaf(e[j], w[j], acc);
    cs[u] = acc;
}

// ---------------------------------------------------------------------------
// Kernel 2: sequential scan, single wave32.
// Lane i (i = lane & 15) holds W_w row i in 16 VGPRs and state element h[i].
// Each substep: publish h through LDS (in-order DS within one wave),
// 16-FMA dot per lane, native v_tanh_f32, damped blend.
// Lanes 16..31 duplicate lanes 0..15 (harmless; only lanes <16 write).
// ---------------------------------------------------------------------------
__global__ void scan_kernel(const float* __restrict__ W_w,
                            const float* __restrict__ xs,
                            const float* __restrict__ cs,
                            float* __restrict__ hs) {
    const int lane = threadIdx.x;
    const int i = lane & 15;

    float w[D];
#pragma unroll
    for (int j = 0; j < D; ++j) w[j] = W_w[i * D + j];

    __shared__ __align__(16) float sh[D];
    float hd = 0.0f;  // distributed state: this lane's h[i]

    for (int t = 0; t < T_LEN; ++t) {
        hd += xs[t * D + i];              // token injection
        const float c = cs[t * D + i];    // W_b[i] + (x @ Wx_w.T)[i], fixed per t
#pragma unroll
        for (int s = 0; s < 2; ++s) {     // RELAX_STEPS
            __syncthreads();              // single-wave: ~free; compiler fence
            if (lane < 16) sh[i] = hd;
            __syncthreads();
            float acc = c;
#pragma unroll
            for (int j = 0; j < D; ++j) acc = fmaf(w[j], sh[j], acc);
            float th;
            // native CDNA5 transcendental; v_nop covers the trans-op
            // result-use hazard (1 independent op required after trans)
            asm volatile("v_tanh_f32 %0, %1\n\tv_nop"
                         : "=v"(th) : "v"(acc));
            hd = hd + ALPHA * (th - hd);  // damped relaxation blend
        }
        if (lane < 16) hs[t * D + i] = hd;
    }
}

// ---------------------------------------------------------------------------
// Kernel 3: logits = hs @ out_w.T + out_b using V_WMMA_F32_16X16X4_F32.
// One wave per 16-col N-tile; B tile (out_w) + bias cached in registers,
// loop over 16 M-tiles (256 rows). K=16 -> 4 chained WMMAs per tile.
//
// Layout (ISA 7.12.2, f32): lane L -> n = L%16, g = L>>4.
//   A elem v of chunk kc : hs[m0 + L%16][4*kc + 2*g + v]
//   B elem v of chunk kc : out_w[n0 + L%16][4*kc + 2*g + v]   (B[k][n]=out_w[n][k])
//   D elem v             : row m0 + v + 8*g, col n0 + L%16
//
// Bias is folded into the C accumulator (column-constant per lane).
// Store offsets (v+8g)*V + n are loop-invariant per lane; only a uniform
// base pointer advances per M-tile -> saddr+voffset stores, minimal VALU.
// NT temporal hint on stores: output is write-once, never re-read.
// EXEC all-ones through the WMMAs; vocab edge via clamped loads + predicated
// stores after the matrix ops.
// ---------------------------------------------------------------------------
__global__ void logits_kernel(const float* __restrict__ hs,
                              const float* __restrict__ out_w,
                              const float* __restrict__ out_b,
                              float* __restrict__ out) {
    const int lane = threadIdx.x;   // 0..31
    const int nl = lane & 15;
    const int g  = lane >> 4;

    const int n0 = blockIdx.x * 16;
    const int n  = n0 + nl;
    const int nc = (n < VOCABN) ? n : (VOCABN - 1);  // clamp for loads

    // B tile: registers for the whole kernel
    v2f b[4];
#pragma unroll
    for (int kc = 0; kc < 4; ++kc) {
        const float* p = out_w + nc * D + kc * 4 + g * 2;
        b[kc][0] = p[0];
        b[kc][1] = p[1];
    }
    const float bias = out_b[nc];

    // per-lane store offsets, fixed across M-tiles
    int soff[8];
#pragma unroll
    for (int v = 0; v < 8; ++v) soff[v] = (v + 8 * g) * VOCABN + n;

    const int mtile0 = blockIdx.y * 16;
    // uniform base pointer, advanced by 16 rows per iteration
    float* base = out + (long long)(mtile0 * 16) * VOCABN;
    const long long step = 16LL * VOCABN;

    const float* arow = hs + (long long)(mtile0 * 16 + nl) * D;

    for (int mt = 0; mt < 16; ++mt) {
        v2f a[4];
#pragma unroll
        for (int kc = 0; kc < 4; ++kc) {
            const float* p = arow + kc * 4 + g * 2;
            a[kc][0] = p[0];
            a[kc][1] = p[1];
        }

        // C pre-loaded with bias: D = A x B + bias
        v8f c;
#pragma unroll
        for (int v = 0; v < 8; ++v) c[v] = bias;

#pragma unroll
        for (int kc = 0; kc < 4; ++kc) {
            // (neg_a, A, neg_b, B, c_mod, C, reuse_a, reuse_b)
            c = __builtin_amdgcn_wmma_f32_16x16x4_f32(
                    false, a[kc], false, b[kc], (short)0, c, false, false);
        }

        if (n < VOCABN) {
#pragma unroll
            for (int v = 0; v < 8; ++v) {
                __builtin_nontemporal_store(c[v], base + soff[v]);
            }
        }
        base += step;
        arow += 16 * D;
    }
}

// ---------------------------------------------------------------------------
extern "C" void kernel_launch(void* const* d_in, const int* in_sizes, int n_in,
                              void* d_out, int out_size, void* d_ws, size_t ws_size,
                              hipStream_t stream) {
    const int*   tok   = (const int*)d_in[0];
    const float* embed = (const float*)d_in[1];
    const float* W_w   = (const float*)d_in[2];
    const float* W_b   = (const float*)d_in[3];
    const float* Wx_w  = (const float*)d_in[4];
    const float* out_w = (const float*)d_in[5];
    const float* out_b = (const float*)d_in[6];
    float* out = (float*)d_out;

    // workspace: xs | cs | hs, each T*D floats (256 KB each)
    float* xs = (float*)d_ws;
    float* cs = xs + T_LEN * D;
    float* hs = cs + T_LEN * D;

    precompute_kernel<<<(T_LEN * D + 255) / 256, 256, 0, stream>>>(
        tok, embed, Wx_w, W_b, xs, cs);

    scan_kernel<<<1, 32, 0, stream>>>(W_w, xs, cs, hs);

    logits_kernel<<<dim3((VOCABN + 15) / 16, T_LEN / 256), 32, 0, stream>>>(
        hs, out_w, out_b, out);
}